// Differentiable_Global_Geometry_PointCloud_22797686407171
// MI455X (gfx1250) — compile-verified
//
#include <hip/hip_runtime.h>
#include <math.h>

// ---------------------------------------------------------------------------
// Differentiable global geometry of a point cloud -> Euler characteristic.
// points: (1, 4096, 3) fp32, k = 50, local_W = 64.
//
// Pipeline (5 kernels):
//   1) topk_kernel     : per-point kNN via LDS distance buffer + iterative argmin
//   2) frames_kernel   : 3x3 covariance + closed-form symmetric eigensolver
//   3) geom_kernel     : tangent projections, bbox, coords, 2x2 curvature -> gauss
//   4) voronoi_kernel  : WMMA f32 16x16x4 grid.e_j GEMM; ballot_w32 epilogue
//                        (per-cell AND over columns becomes 2 SALU compares per
//                        D register -> co-executes with the WMMA pipe)
//   5) reduce_kernel   : weighted sum -> euler scalar
//
// Working set ~2.6MB: fully L2-resident; compute-bound on the Voronoi pass,
// which is mapped onto V_WMMA_F32_16X16X4_F32 (wave32).
// ---------------------------------------------------------------------------

typedef float v2f __attribute__((ext_vector_type(2)));
typedef float v8f __attribute__((ext_vector_type(8)));

#define KNN 50
#define LOCAL_W 64
#define NCELLS (LOCAL_W * LOCAL_W)   // 4096 grid cells
#define MAXN 4096

// ---------------------------------------------------------------------------
// Kernel 1: per-point kNN. One workgroup per point; full distance row in LDS,
// then KNN iterations of (min,index) tree reduction with lowest-index ties
// (matches jax.lax.top_k ordering on -dist).
// ---------------------------------------------------------------------------
__global__ __launch_bounds__(256) void topk_kernel(const float* __restrict__ pts,
                                                   int* __restrict__ idxout, int n) {
    __shared__ float dist[MAXN];
    __shared__ float sval[256];
    __shared__ int   sidx[256];
    const int i   = blockIdx.x;
    const int tid = threadIdx.x;
    const float px = pts[i * 3 + 0], py = pts[i * 3 + 1], pz = pts[i * 3 + 2];
    for (int j = tid; j < n; j += 256) {
        float dx = pts[j * 3 + 0] - px;
        float dy = pts[j * 3 + 1] - py;
        float dz = pts[j * 3 + 2] - pz;
        dist[j] = dx * dx + dy * dy + dz * dz;
    }
    __syncthreads();
    for (int it = 0; it < KNN; ++it) {
        float bv = 3.4e38f; int bi = n;
        for (int j = tid; j < n; j += 256) {
            float v = dist[j];
            if (v < bv) { bv = v; bi = j; }
        }
        sval[tid] = bv; sidx[tid] = bi;
        __syncthreads();
        for (int s = 128; s > 0; s >>= 1) {
            if (tid < s) {
                float v2 = sval[tid + s]; int i2 = sidx[tid + s];
                if (v2 < sval[tid] || (v2 == sval[tid] && i2 < sidx[tid])) {
                    sval[tid] = v2; sidx[tid] = i2;
                }
            }
            __syncthreads();
        }
        if (tid == 0) { idxout[i * KNN + it] = sidx[0]; dist[sidx[0]] = 3.4e38f; }
        __syncthreads();
    }
}

// ---------------------------------------------------------------------------
// Closed-form symmetric 3x3 eigensolver helpers (Smith's method).
// ---------------------------------------------------------------------------
__device__ inline void cross3(float ax, float ay, float az,
                              float bx, float by, float bz,
                              float& cx, float& cy, float& cz) {
    cx = ay * bz - az * by;
    cy = az * bx - ax * bz;
    cz = ax * by - ay * bx;
}

__device__ inline void eigvec3(float a00, float a01, float a02,
                               float a11, float a12, float a22, float lam,
                               float& vx, float& vy, float& vz) {
    float r0x = a00 - lam, r0y = a01,       r0z = a02;
    float r1x = a01,       r1y = a11 - lam, r1z = a12;
    float r2x = a02,       r2y = a12,       r2z = a22 - lam;
    float c0x, c0y, c0z, c1x, c1y, c1z, c2x, c2y, c2z;
    cross3(r0x, r0y, r0z, r1x, r1y, r1z, c0x, c0y, c0z);
    cross3(r0x, r0y, r0z, r2x, r2y, r2z, c1x, c1y, c1z);
    cross3(r1x, r1y, r1z, r2x, r2y, r2z, c2x, c2y, c2z);
    float n0 = c0x * c0x + c0y * c0y + c0z * c0z;
    float n1 = c1x * c1x + c1y * c1y + c1z * c1z;
    float n2 = c2x * c2x + c2y * c2y + c2z * c2z;
    float bx = c0x, by = c0y, bz = c0z, bn = n0;
    if (n1 > bn) { bx = c1x; by = c1y; bz = c1z; bn = n1; }
    if (n2 > bn) { bx = c2x; by = c2y; bz = c2z; bn = n2; }
    if (bn < 1e-30f) { vx = 1.f; vy = 0.f; vz = 0.f; return; }
    float inv = rsqrtf(bn);
    vx = bx * inv; vy = by * inv; vz = bz * inv;
}

// ---------------------------------------------------------------------------
// Kernel 2: covariance of kNN + eigenframes (rows: normal, t1, t2), det fix.
// ---------------------------------------------------------------------------
__global__ __launch_bounds__(256) void frames_kernel(const float* __restrict__ pts,
                                                     const int* __restrict__ idx,
                                                     float* __restrict__ frames, int n) {
    int i = blockIdx.x * blockDim.x + threadIdx.x;
    if (i >= n) return;
    float sx = 0, sy = 0, sz = 0;
    float xx = 0, xy = 0, xz = 0, yy = 0, yz = 0, zz = 0;
    for (int j = 0; j < KNN; ++j) {
        int p = idx[i * KNN + j];
        float x = pts[p * 3 + 0], y = pts[p * 3 + 1], z = pts[p * 3 + 2];
        sx += x; sy += y; sz += z;
        xx += x * x; xy += x * y; xz += x * z;
        yy += y * y; yz += y * z; zz += z * z;
    }
    const float inv = 1.f / (float)KNN;
    // covs = (sum p p^T - s s^T / k) / 2    (reference divides by 2)
    float a00 = (xx - sx * sx * inv) * 0.5f;
    float a01 = (xy - sx * sy * inv) * 0.5f;
    float a02 = (xz - sx * sz * inv) * 0.5f;
    float a11 = (yy - sy * sy * inv) * 0.5f;
    float a12 = (yz - sy * sz * inv) * 0.5f;
    float a22 = (zz - sz * sz * inv) * 0.5f;

    float v0x = 1, v0y = 0, v0z = 0;   // eigenvector of smallest eigenvalue
    float v2x = 0, v2y = 0, v2z = 1;   // eigenvector of largest eigenvalue
    float q  = (a00 + a11 + a22) * (1.f / 3.f);
    float p1 = a01 * a01 + a02 * a02 + a12 * a12;
    float p2 = (a00 - q) * (a00 - q) + (a11 - q) * (a11 - q) + (a22 - q) * (a22 - q) + 2.f * p1;
    if (p2 > 1e-24f) {
        float p  = sqrtf(p2 * (1.f / 6.f));
        float ip = 1.f / p;
        float b00 = (a00 - q) * ip, b01 = a01 * ip, b02 = a02 * ip;
        float b11 = (a11 - q) * ip, b12 = a12 * ip, b22 = (a22 - q) * ip;
        float detB = b00 * (b11 * b22 - b12 * b12)
                   - b01 * (b01 * b22 - b12 * b02)
                   + b02 * (b01 * b12 - b11 * b02);
        float r = fminf(fmaxf(detB * 0.5f, -1.f), 1.f);
        float phi = acosf(r) * (1.f / 3.f);
        float eMax = q + 2.f * p * cosf(phi);
        float eMin = q + 2.f * p * cosf(phi + 2.0943951023931953f); // +2pi/3
        eigvec3(a00, a01, a02, a11, a12, a22, eMin, v0x, v0y, v0z);
        eigvec3(a00, a01, a02, a11, a12, a22, eMax, v2x, v2y, v2z);
    }
    // middle eigenvector = v2 x v0 (orthonormal completion)
    float v1x, v1y, v1z;
    cross3(v2x, v2y, v2z, v0x, v0y, v0z, v1x, v1y, v1z);
    float nn = v1x * v1x + v1y * v1y + v1z * v1z;
    if (nn < 1e-30f) { v1x = 0.f; v1y = 1.f; v1z = 0.f; }
    else { float s = rsqrtf(nn); v1x *= s; v1y *= s; v1z *= s; }
    // det of frame (rows v0,v1,v2); multiply row1 by det (reference det fix)
    float cx, cy, cz;
    cross3(v1x, v1y, v1z, v2x, v2y, v2z, cx, cy, cz);
    float det = v0x * cx + v0y * cy + v0z * cz;
    v1x *= det; v1y *= det; v1z *= det;
    float* f = frames + (size_t)i * 9;
    f[0] = v0x; f[1] = v0y; f[2] = v0z;
    f[3] = v1x; f[4] = v1y; f[5] = v1z;
    f[6] = v2x; f[7] = v2y; f[8] = v2z;
}

// ---------------------------------------------------------------------------
// Kernel 3: tangent projections dt/dn, bbox -> coords, 2x2 curvature -> gauss.
// gauss = det(Q E Q^T) = det(E) since det(Q)^2 = 1.
// ---------------------------------------------------------------------------
__global__ __launch_bounds__(256) void geom_kernel(const float* __restrict__ pts,
                                                   const int* __restrict__ idx,
                                                   const float* __restrict__ frames,
                                                   float* __restrict__ coords,
                                                   float* __restrict__ maxlen,
                                                   float* __restrict__ gauss, int n) {
    int i = blockIdx.x * blockDim.x + threadIdx.x;
    if (i >= n) return;
    const float* f = frames + (size_t)i * 9;
    float n0x = f[0], n0y = f[1], n0z = f[2];
    float t1x = f[3], t1y = f[4], t1z = f[5];
    float t2x = f[6], t2y = f[7], t2z = f[8];
    float px = pts[i * 3 + 0], py = pts[i * 3 + 1], pz = pts[i * 3 + 2];

    float mnx = 3.4e38f, mny = 3.4e38f, mxx = -3.4e38f, mxy = -3.4e38f;
    float sxx = 0, sxy = 0, syy = 0;                 // XXT
    float y00 = 0, y01 = 0, y10 = 0, y11 = 0;        // YXT
    for (int j = 0; j < KNN; ++j) {
        int p = idx[i * KNN + j];
        float dx = pts[p * 3 + 0] - px;
        float dy = pts[p * 3 + 1] - py;
        float dz = pts[p * 3 + 2] - pz;
        float u = dx * t1x + dy * t1y + dz * t1z;   // dt[...,0]
        float v = dx * t2x + dy * t2y + dz * t2z;   // dt[...,1]
        const float* fp = frames + (size_t)p * 9;
        float lx = fp[0] - n0x, ly = fp[1] - n0y, lz = fp[2] - n0z;
        float a = lx * t1x + ly * t1y + lz * t1z;   // dn[...,0]
        float b = lx * t2x + ly * t2y + lz * t2z;   // dn[...,1]
        mnx = fminf(mnx, u); mny = fminf(mny, v);
        mxx = fmaxf(mxx, u); mxy = fmaxf(mxy, v);
        sxx += u * u; sxy += u * v; syy += v * v;
        y00 += a * u; y01 += a * v; y10 += b * u; y11 += b * v;
    }
    const float rescale = 1.1f;
    float bminx = mnx * rescale, bminy = mny * rescale;
    float bmaxx = mxx * rescale, bmaxy = mxy * rescale;
    float ml = fmaxf(bmaxx - bminx, bmaxy - bminy);
    maxlen[i] = ml;
    float s2 = 2.f / fmaxf(ml, 1e-20f);
    // second pass: write normalized coords (recompute the gathers; L2-resident)
    for (int j = 0; j < KNN; ++j) {
        int p = idx[i * KNN + j];
        float dx = pts[p * 3 + 0] - px;
        float dy = pts[p * 3 + 1] - py;
        float dz = pts[p * 3 + 2] - pz;
        float u = dx * t1x + dy * t1y + dz * t1z;
        float v = dx * t2x + dy * t2y + dz * t2z;
        coords[((size_t)i * KNN + j) * 2 + 0] = (u - bminx) * s2 - 1.f;
        coords[((size_t)i * KNN + j) * 2 + 1] = (v - bminy) * s2 - 1.f;
    }
    // S = YXT + YXT^T
    float S00 = 2.f * y00, S01 = y01 + y10, S11 = 2.f * y11;
    // 2x2 symmetric eigh of XXT (ascending a <= b)
    float tr = sxx + syy, df = sxx - syy;
    float sq = sqrtf(df * df + 4.f * sxy * sxy);
    float a = 0.5f * (tr - sq), b = 0.5f * (tr + sq);
    float v1x = sxy,      v1y = a - sxx;
    float w1x = a - syy,  w1y = sxy;
    if (v1x * v1x + v1y * v1y < w1x * w1x + w1y * w1y) { v1x = w1x; v1y = w1y; }
    float nn = v1x * v1x + v1y * v1y;
    float q1x, q1y;
    if (nn < 1e-30f) { q1x = 1.f; q1y = 0.f; }
    else { float is = rsqrtf(nn); q1x = v1x * is; q1y = v1y * is; }
    float q2x = -q1y, q2y = q1x;
    // QTSQ = Q^T S Q
    float sq00 = S00 * q1x + S01 * q1y, sq10 = S01 * q1x + S11 * q1y;
    float sq01 = S00 * q2x + S01 * q2y, sq11 = S01 * q2x + S11 * q2y;
    float m00 = q1x * sq00 + q1y * sq10;
    float m01 = q1x * sq01 + q1y * sq11;
    float m10 = q2x * sq00 + q2y * sq10;
    float m11 = q2x * sq01 + q2y * sq11;
    float e00 = m00 / (2.f * a + 1e-8f);
    float e01 = m01 / (a + b + 1e-8f);
    float e10 = m10 / (a + b + 1e-8f);
    float e11 = m11 / (2.f * b + 1e-8f);
    gauss[i] = e00 * e11 - e01 * e10;   // det(Wmap) == det(E)
}

// ---------------------------------------------------------------------------
// Kernel 4: Voronoi cell counting with WMMA f32 16x16x4.
// Cell g is counted for point i iff for all j: g.e_j <= b_j, where
// e_j = c_j - c_0 and b_j = (|c_j|^2 - |c_0|^2)/2 (j padded to 64 with e=0,b=0).
// A-frag (16 cells x K): lanes 0-15 hold (gx,gy) [K=0,1]; lanes 16-31 zeros
// [K=2,3]; B-frag mirrors with e-vectors.  D layout: VGPR r -> M = r
// (lanes 0-15, N=lane) / M = 8+r (lanes 16-31, N=lane-16).
// Epilogue: ballot_w32(d[r] <= b) gives, per D register, cell r's verdict in
// bits[15:0] and cell 8+r's in bits[31:16] -- two scalar compares per register,
// all-SALU mask accumulation that co-executes with the WMMA pipe.
// ---------------------------------------------------------------------------
__global__ __launch_bounds__(128) void voronoi_kernel(const float* __restrict__ coords,
                                                      int* __restrict__ counts, int n) {
    __shared__ float exs[64], eys[64], bbs[64];
    __shared__ int cnt;
    const int i   = blockIdx.x;
    const int tid = threadIdx.x;
    if (tid == 0) cnt = 0;
    const float c0x = coords[(size_t)i * KNN * 2 + 0];
    const float c0y = coords[(size_t)i * KNN * 2 + 1];
    if (tid < 64) {
        int j = (tid < KNN) ? tid : 0;     // pad -> e=0, b=0 (always satisfied)
        float cx = coords[((size_t)i * KNN + j) * 2 + 0];
        float cy = coords[((size_t)i * KNN + j) * 2 + 1];
        exs[tid] = cx - c0x;
        eys[tid] = cy - c0y;
        bbs[tid] = 0.5f * ((cx * cx + cy * cy) - (c0x * c0x + c0y * c0y));
    }
    __syncthreads();

    const int lane = tid & 31;
    const int wave = tid >> 5;
    const bool hi  = lane >= 16;
    const int  l   = lane & 15;

    v2f  Bf[4];
    float bv[4];
#pragma unroll
    for (int t = 0; t < 4; ++t) {
        int j = t * 16 + l;
        Bf[t][0] = hi ? 0.f : exs[j];
        Bf[t][1] = hi ? 0.f : eys[j];
        bv[t]    = bbs[j];                 // column n = lane%16 in both halves
    }

    const float step = 2.f / (float)(LOCAL_W - 1);
    int wcount = 0;
    for (int mt = wave; mt < NCELLS / 16; mt += 4) {      // uniform across wave
        int cell = mt * 16 + l;
        float gx = -1.f + step * (float)(cell >> 6);
        float gy = -1.f + step * (float)(cell & 63);
        v2f Af;
        Af[0] = hi ? 0.f : gx;
        Af[1] = hi ? 0.f : gy;
        unsigned ok16 = 0xFFFFu;
#pragma unroll
        for (int t = 0; t < 4; ++t) {
            v8f c = {0.f, 0.f, 0.f, 0.f, 0.f, 0.f, 0.f, 0.f};
            v8f d = __builtin_amdgcn_wmma_f32_16x16x4_f32(
                false, Af, false, Bf[t], (short)0, c, false, false);
            unsigned okt = 0;
#pragma unroll
            for (int r = 0; r < 8; ++r) {
                unsigned m = __builtin_amdgcn_ballot_w32(d[r] <= bv[t]);
                okt |= ((m & 0xFFFFu) == 0xFFFFu) ? (1u << r) : 0u;        // cell r
                okt |= ((m >> 16) == 0xFFFFu) ? (1u << (8 + r)) : 0u;      // cell 8+r
            }
            ok16 &= okt;
        }
        wcount += __popc(ok16);      // uniform across the wave (SALU-derived)
    }
    if (lane == 0) atomicAdd(&cnt, wcount);
    __syncthreads();
    if (tid == 0) counts[i] = cnt;
}

// ---------------------------------------------------------------------------
// Kernel 5: euler = sum(gauss * counts * max_len^2) / (W-1)^2 / (2*pi)
// ---------------------------------------------------------------------------
__global__ __launch_bounds__(256) void reduce_kernel(const float* __restrict__ gauss,
                                                     const int* __restrict__ counts,
                                                     const float* __restrict__ maxlen,
                                                     float* __restrict__ out, int n) {
    __shared__ float s[256];
    float acc = 0.f;
    for (int i = threadIdx.x; i < n; i += 256) {
        float ml = maxlen[i];
        acc += gauss[i] * (float)counts[i] * ml * ml;
    }
    s[threadIdx.x] = acc;
    __syncthreads();
    for (int st = 128; st > 0; st >>= 1) {
        if (threadIdx.x < st) s[threadIdx.x] += s[threadIdx.x + st];
        __syncthreads();
    }
    if (threadIdx.x == 0) {
        const float invW2 = 1.f / ((float)(LOCAL_W - 1) * (float)(LOCAL_W - 1));
        out[0] = s[0] * invW2 / (2.f * 3.14159265358979323846f);
    }
}

// ---------------------------------------------------------------------------
extern "C" void kernel_launch(void* const* d_in, const int* in_sizes, int n_in,
                              void* d_out, int out_size, void* d_ws, size_t ws_size,
                              hipStream_t stream) {
    const float* pts = (const float*)d_in[0];
    const int n = in_sizes[0] / 3;   // B*N points (B=1, N=4096 in setup)

    char* ws = (char*)d_ws;
    size_t off = 0;
    int*   idx    = (int*)(ws + off);   off += (size_t)n * KNN * sizeof(int);
    float* frames = (float*)(ws + off); off += (size_t)n * 9 * sizeof(float);
    float* coords = (float*)(ws + off); off += (size_t)n * KNN * 2 * sizeof(float);
    float* maxlen = (float*)(ws + off); off += (size_t)n * sizeof(float);
    float* gauss  = (float*)(ws + off); off += (size_t)n * sizeof(float);
    int*   counts = (int*)(ws + off);   off += (size_t)n * sizeof(int);

    topk_kernel  <<<n, 256, 0, stream>>>(pts, idx, n);
    frames_kernel<<<(n + 255) / 256, 256, 0, stream>>>(pts, idx, frames, n);
    geom_kernel  <<<(n + 255) / 256, 256, 0, stream>>>(pts, idx, frames, coords,
                                                       maxlen, gauss, n);
    voronoi_kernel<<<n, 128, 0, stream>>>(coords, counts, n);
    reduce_kernel<<<1, 256, 0, stream>>>(gauss, counts, maxlen, (float*)d_out, n);
}